// MemoryWeightedAttention_8108898255658
// MI455X (gfx1250) — compile-verified
//
#include <hip/hip_runtime.h>
#include <hip/hip_bf16.h>
#include <math.h>

typedef __attribute__((ext_vector_type(2))) float v2f;
typedef __attribute__((ext_vector_type(4))) float f4;
typedef __attribute__((ext_vector_type(8))) float v8f;

#define S_LEN 2048
#define VD    288
#define HD2   576     // 2*VD
#define NH    16
#define FFN   512
#define FFNH  256
#define CTX   1024
#define EPSILON 0.066f
#define LN_EPS  1e-5f

// ---------------- wave32 helpers ----------------

__device__ __forceinline__ float wave_reduce_sum(float x) {
#pragma unroll
  for (int off = 16; off > 0; off >>= 1)
    x += __shfl_down(x, off, 32);
  return __shfl(x, 0, 32);
}

__device__ __forceinline__ float gelu_exact(float x) {
  return 0.5f * x * (1.0f + erff(x * 0.70710678118654752f));
}

// One wave computes one 16(M) x 16(N) fp32 tile:
//   C = A[16 x Kpad] (LDS, row-major, stride ldA) * B[K x ldB] (global, row-major)
// using V_WMMA_F32_16X16X4_F32.  B rows >= Klimit read as zero (K padding).
// ISA layouts (CDNA5 ISA):
//   A 16x4 : lane m=lane&15, half h=lane>>4; VGPR0=K(2h), VGPR1=K(2h+1)
//   B 4x16 : rows striped across lanes; VGPR r, half h -> K = r + 2h, N = lane&15
//   C/D    : VGPR r, half h -> M = r + 8h, N = lane&15
__device__ __forceinline__ v8f wmma_tile(const float* As, int ldA,
                                         const float* __restrict__ Bg, int ldB,
                                         int n0, int Kpad, int Klimit, int lane) {
  const int m = lane & 15;
  const int h = lane >> 4;
  v8f acc = {};
  for (int k = 0; k < Kpad; k += 4) {
    const int ka = k + 2 * h;
    v2f a, b;
    a.x = As[m * ldA + ka];
    a.y = As[m * ldA + ka + 1];
    b.x = (ka     < Klimit) ? Bg[(size_t)ka       * ldB + n0 + m] : 0.0f;
    b.y = (ka + 1 < Klimit) ? Bg[(size_t)(ka + 1) * ldB + n0 + m] : 0.0f;
    acc = __builtin_amdgcn_wmma_f32_16x16x4_f32(false, a, false, b,
                                                (short)0, acc, false, false);
  }
  return acc;
}

// act: 0 = gelu(exact), 1 = tanh
__device__ __forceinline__ void store_tile(float* Ds, int ldD, int n0, v8f acc,
                                           const float* __restrict__ bias,
                                           int lane, int act) {
  const int m = lane & 15;
  const int h = lane >> 4;
  const float bv = bias[n0 + m];
#pragma unroll
  for (int r = 0; r < 8; ++r) {
    float x = acc[r] + bv;
    x = (act == 0) ? gelu_exact(x) : tanhf(x);
    Ds[(r + 8 * h) * ldD + n0 + m] = x;
  }
}

// ---------------- kernels ----------------

__global__ void zero_counts_kernel(int* __restrict__ counts) {
  int i = blockIdx.x * blockDim.x + threadIdx.x;
  if (i < CTX) counts[i] = 0;
}

__global__ void hist_kernel(const int* __restrict__ ids, int* __restrict__ counts, int n) {
  int i = blockIdx.x * blockDim.x + threadIdx.x;
  if (i < n) atomicAdd(&counts[ids[i]], 1);
}

// One workgroup (256 thr = 8 waves) handles 16 rows (b,s) end-to-end -> mod[row].
__global__ __launch_bounds__(256)
void fuse_mod_kernel(const int* __restrict__ ids,
                     const float* __restrict__ embW,
                     const float* __restrict__ ln_v_w, const float* __restrict__ ln_v_b,
                     const float* __restrict__ pW1, const float* __restrict__ pb1,
                     const float* __restrict__ pW2, const float* __restrict__ pb2,
                     const float* __restrict__ ln_c_w, const float* __restrict__ ln_c_b,
                     const float* __restrict__ fW1, const float* __restrict__ fb1,
                     const float* __restrict__ fW2, const float* __restrict__ fb2,
                     const float* __restrict__ fW3, const float* __restrict__ fb3,
                     const int* __restrict__ counts,
                     float* __restrict__ modArr) {
  __shared__ float Av[16 * VD];       // v (16x288); reused as h3 (16x256)
  __shared__ float Bh[16 * HD2];      // h1 (16x576); reused as h2 (16x512)
  __shared__ float Ccomb[16 * 20];    // comb, K=17 zero-padded to 20
  __shared__ float valTile[16 * NH];  // tanh valence 16x16
  __shared__ float occs[16];

  const int tid  = threadIdx.x;
  const int lane = tid & 31;
  const int wave = tid >> 5;
  const int row0 = blockIdx.x * 16;

  // ---- stage 0: gather embedding rows + occurrence feature ----
  {
    const int mr = tid >> 4;   // 0..15
    const int jc = tid & 15;
    const int tok = ids[row0 + mr];
    const float* src = embW + (size_t)tok * VD;
    for (int j = jc; j < VD; j += 16)
      Av[mr * VD + j] = src[j];
    if (jc == 0)
      occs[mr] = log1pf(fmaxf((float)counts[tok], 1.0f));
  }
  __syncthreads();

  // ---- stage 1: layernorm v (two-pass, wave per row) ----
  for (int mr = wave; mr < 16; mr += 8) {
    float* rowp = Av + mr * VD;
    float s = 0.f;
    for (int j = lane; j < VD; j += 32) s += rowp[j];
    s = wave_reduce_sum(s);
    const float mean = s * (1.0f / VD);
    float s2 = 0.f;
    for (int j = lane; j < VD; j += 32) { float d = rowp[j] - mean; s2 += d * d; }
    s2 = wave_reduce_sum(s2);
    const float inv = rsqrtf(s2 * (1.0f / VD) + LN_EPS);
    for (int j = lane; j < VD; j += 32)
      rowp[j] = (rowp[j] - mean) * inv * ln_v_w[j] + ln_v_b[j];
  }
  __syncthreads();

  // ---- stage 2: h1 = gelu(v @ pW1 + pb1)   [16x288]x[288x576] ----
  for (int nt = wave; nt < HD2 / 16; nt += 8) {
    v8f acc = wmma_tile(Av, VD, pW1, HD2, nt * 16, VD, VD, lane);
    store_tile(Bh, HD2, nt * 16, acc, pb1, lane, 0);
  }
  __syncthreads();

  // ---- stage 3: valence = tanh(h1 @ pW2 + pb2)  [16x576]x[576x16] ----
  if (wave == 0) {
    v8f acc = wmma_tile(Bh, HD2, pW2, NH, 0, HD2, HD2, lane);
    store_tile(valTile, NH, 0, acc, pb2, lane, 1);
  }
  __syncthreads();

  // ---- stage 4: comb = layernorm(concat(occ, valence)), zero-pad K 17->20 ----
  if (tid < 16) {
    float vals[17];
    vals[0] = occs[tid];
#pragma unroll
    for (int j = 0; j < 16; ++j) vals[1 + j] = valTile[tid * NH + j];
    float s = 0.f;
#pragma unroll
    for (int j = 0; j < 17; ++j) s += vals[j];
    const float mean = s * (1.0f / 17.0f);
    float s2 = 0.f;
#pragma unroll
    for (int j = 0; j < 17; ++j) { float d = vals[j] - mean; s2 += d * d; }
    const float inv = rsqrtf(s2 * (1.0f / 17.0f) + LN_EPS);
#pragma unroll
    for (int j = 0; j < 17; ++j)
      Ccomb[tid * 20 + j] = (vals[j] - mean) * inv * ln_c_w[j] + ln_c_b[j];
    Ccomb[tid * 20 + 17] = 0.f;
    Ccomb[tid * 20 + 18] = 0.f;
    Ccomb[tid * 20 + 19] = 0.f;
  }
  __syncthreads();

  // ---- stage 5: h2 = gelu(comb @ fW1 + fb1)  K=17 (pad 20), N=512 ----
  for (int nt = wave; nt < FFN / 16; nt += 8) {
    v8f acc = wmma_tile(Ccomb, 20, fW1, FFN, nt * 16, 20, 17, lane);
    store_tile(Bh, FFN, nt * 16, acc, fb1, lane, 0);
  }
  __syncthreads();

  // ---- stage 6: h3 = gelu(h2 @ fW2 + fb2)  [16x512]x[512x256] ----
  for (int nt = wave; nt < FFNH / 16; nt += 8) {
    v8f acc = wmma_tile(Bh, FFN, fW2, FFNH, nt * 16, FFN, FFN, lane);
    store_tile(Av, FFNH, nt * 16, acc, fb2, lane, 0);
  }
  __syncthreads();

  // ---- stage 7: mod = tanh(h3 @ fW3 + fb3)  (N=1, wave reduction) ----
  for (int mr = wave; mr < 16; mr += 8) {
    float s = 0.f;
    for (int j = lane; j < FFNH; j += 32) s += Av[mr * FFNH + j] * fW3[j];
    s = wave_reduce_sum(s);
    if (lane == 0) modArr[row0 + mr] = tanhf(s + fb3[0]);
  }
}

// out[b,h,sq,sk] = scores * (1 + eps*mod[b,sq]); 512MB stream -> nontemporal.
__global__ __launch_bounds__(256)
void scale_kernel(const f4* __restrict__ in, f4* __restrict__ out,
                  const float* __restrict__ modArr) {
  const size_t t = (size_t)blockIdx.x * blockDim.x + threadIdx.x;
  const size_t flat = t << 2;                 // element index
  const int b  = (int)(flat >> 26);           // H*S*S = 2^26
  const int sq = (int)((flat >> 11) & 2047);  // S = 2^11
  const float w = 1.0f + EPSILON * modArr[(b << 11) + sq];
  f4 v = __builtin_nontemporal_load(&in[t]);
  v.x *= w; v.y *= w; v.z *= w; v.w *= w;
  __builtin_nontemporal_store(v, &out[t]);
}

// ---------------- launch ----------------

extern "C" void kernel_launch(void* const* d_in, const int* in_sizes, int n_in,
                              void* d_out, int out_size, void* d_ws, size_t ws_size,
                              hipStream_t stream) {
  (void)n_in; (void)ws_size;
  const float* attn   = (const float*)d_in[0];
  const int*   ids    = (const int*)d_in[1];
  const float* embW   = (const float*)d_in[2];
  const float* ln_v_w = (const float*)d_in[3];
  const float* ln_v_b = (const float*)d_in[4];
  const float* pW1    = (const float*)d_in[5];
  const float* pb1    = (const float*)d_in[6];
  const float* pW2    = (const float*)d_in[7];
  const float* pb2    = (const float*)d_in[8];
  const float* ln_c_w = (const float*)d_in[9];
  const float* ln_c_b = (const float*)d_in[10];
  const float* fW1    = (const float*)d_in[11];
  const float* fb1    = (const float*)d_in[12];
  const float* fW2    = (const float*)d_in[13];
  const float* fb2    = (const float*)d_in[14];
  const float* fW3    = (const float*)d_in[15];
  const float* fb3    = (const float*)d_in[16];

  int*   counts = (int*)d_ws;
  float* modArr = (float*)((char*)d_ws + CTX * sizeof(int));

  const int nIds = in_sizes[1];               // B*S = 4096

  zero_counts_kernel<<<(CTX + 255) / 256, 256, 0, stream>>>(counts);
  hist_kernel<<<(nIds + 255) / 256, 256, 0, stream>>>(ids, counts, nIds);

  fuse_mod_kernel<<<nIds / 16, 256, 0, stream>>>(
      ids, embW, ln_v_w, ln_v_b, pW1, pb1, pW2, pb2,
      ln_c_w, ln_c_b, fW1, fb1, fW2, fb2, fW3, fb3, counts, modArr);

  const long long total4 = (long long)out_size / 4;   // 33,554,432 float4s
  scale_kernel<<<(unsigned)(total4 / 256), 256, 0, stream>>>(
      (const f4*)attn, (f4*)d_out, modArr);
}